// BatchedRoutingLinear_9869834846657
// MI455X (gfx1250) — compile-verified
//
#include <hip/hip_runtime.h>

typedef __attribute__((ext_vector_type(2))) float v2f;
typedef __attribute__((ext_vector_type(8))) float v8f;

namespace {
constexpr int B_   = 8;
constexpr int IN_  = 512;
constexpr int OUT_ = 128000;
constexpr int TOPK = 32;
constexpr int XSTR = 516;   // LDS row stride in floats: 4*M mod 64 unique for M=0..15
}

// Kernel 1: one wave -> one 16-wide output tile.
// out[b,o] = x[b,:]·w[o,:] ; sim[b,o] = out[b,o] * rsqrt(||w_o||^2)
// (per-row x-norm factor dropped: positive constant per row, ordering-invariant)
__global__ __launch_bounds__(256) void routing_gemm_wmma(
    const float* __restrict__ x, const float* __restrict__ w,
    float* __restrict__ out, float* __restrict__ sim)
{
  __shared__ float xs[16 * XSTR];

  const int tid = threadIdx.x;
  // cooperatively stage x (8x512) into LDS, zero-pad rows 8..15 (WMMA M pad)
  for (int i = tid; i < 16 * IN_; i += 256) {
    const int r = i >> 9;       // / 512
    const int c = i & (IN_ - 1);
    xs[r * XSTR + c] = (r < B_) ? x[r * IN_ + c] : 0.0f;
  }
  __syncthreads();

  const int wave = tid >> 5;
  const int lane = tid & 31;
  const int n0   = (blockIdx.x * 8 + wave) * 16;
  const int nsub = lane & 15;          // N for B-frag, M for A-frag
  const int koff = (lane >> 4) * 2;    // lanes 16..31 hold K+2,K+3

  const float* __restrict__ wrow = w + (size_t)(n0 + nsub) * IN_ + koff;
  const float* __restrict__ xrow = xs + nsub * XSTR + koff;

  v8f acc = {};
  float wn2 = 0.0f;

  #pragma unroll 4
  for (int k = 0; k < IN_; k += 4) {
    v2f a;
    a.x = xrow[k];
    a.y = xrow[k + 1];
    // W is a 262 MB one-shot stream (> 192 MB L2): non-temporal so it doesn't
    // evict the sim/out matrices that kernel 2 re-reads out of L2.
    const v2f b = __builtin_nontemporal_load((const v2f*)(wrow + k));
    wn2 += b.x * b.x + b.y * b.y;
    // 8 args: (neg_a, A, neg_b, B, c_mod, C, reuse_a, reuse_b)
    acc = __builtin_amdgcn_wmma_f32_16x16x4_f32(
        false, a, false, b, (short)0, acc, false, false);
  }

  // column-norm: combine partials across the (lane, lane^16) pair
  wn2 += __shfl_xor(wn2, 16, 32);
  const float inv_norm = __frsqrt_rn(wn2);

  // D layout: VGPR r, lanes 0..15 -> (M=r, N=lane). Only M<8 rows are real.
  if (lane < 16) {
    const int col = n0 + lane;
    #pragma unroll
    for (int r = 0; r < 8; ++r) {
      const float v = acc[r];
      out[(size_t)r * OUT_ + col] = v;
      sim[(size_t)r * OUT_ + col] = v * inv_norm;
    }
  }
}

// Kernel 2: per batch row, 32 rounds of global argmax over the (L2-resident)
// sim row, excluding previously chosen indices, then out[b,idx] += bias[idx].
__global__ __launch_bounds__(1024) void topk_bias_add(
    const float* __restrict__ sim, const float* __restrict__ bias,
    float* __restrict__ out)
{
  __shared__ float rv[1024];
  __shared__ int   ri[1024];
  __shared__ int   chosen[TOPK];

  const int b   = blockIdx.x;
  const int tid = threadIdx.x;
  const float* __restrict__ row = sim + (size_t)b * OUT_;

  for (int t = 0; t < TOPK; ++t) {
    float best = -INFINITY;
    int   bi   = 0x7fffffff;
    for (int i = tid; i < OUT_; i += 1024) {
      bool taken = false;
      for (int j = 0; j < t; ++j) taken |= (chosen[j] == i);
      const float v = taken ? -INFINITY : row[i];
      if (v > best || (v == best && i < bi)) { best = v; bi = i; }
    }
    rv[tid] = best;
    ri[tid] = bi;
    __syncthreads();
    for (int s = 512; s > 0; s >>= 1) {
      if (tid < s) {
        const float ov = rv[tid + s];
        const int   oi = ri[tid + s];
        if (ov > rv[tid] || (ov == rv[tid] && oi < ri[tid])) {
          rv[tid] = ov;
          ri[tid] = oi;
        }
      }
      __syncthreads();
    }
    if (tid == 0) {
      const int idx = ri[0];
      chosen[t] = idx;
      out[(size_t)b * OUT_ + idx] += bias[idx];
    }
    __syncthreads();
  }
}

extern "C" void kernel_launch(void* const* d_in, const int* in_sizes, int n_in,
                              void* d_out, int out_size, void* d_ws, size_t ws_size,
                              hipStream_t stream) {
  const float* x    = (const float*)d_in[0];   // [8, 512]
  const float* w    = (const float*)d_in[1];   // [128000, 512]
  const float* bias = (const float*)d_in[2];   // [128000]
  // d_in[3] = top_k scalar (32), compile-time constant here
  float* out = (float*)d_out;                  // [8, 128000]
  float* sim = (float*)d_ws;                   // 8*128000*4 = 4.1 MB scratch

  routing_gemm_wmma<<<OUT_ / (8 * 16), 256, 0, stream>>>(x, w, out, sim);
  topk_bias_add<<<B_, 1024, 0, stream>>>(sim, bias, out);
}